// VirtualNodeLayer_81784767250590
// MI455X (gfx1250) — compile-verified
//
#include <hip/hip_runtime.h>
#include <hip/hip_bf16.h>

typedef __attribute__((ext_vector_type(2))) float v2f;
typedef __attribute__((ext_vector_type(4))) float v4f;
typedef __attribute__((ext_vector_type(8))) float v8f;

#define H 128
#define H2 256

// ---------------------------------------------------------------------------
// K0: zero the accumulator region of the workspace (seg_sum + counts)
// ---------------------------------------------------------------------------
__global__ __launch_bounds__(256) void zero_kernel(float* p, int n) {
    int i = blockIdx.x * 256 + threadIdx.x;
    if (i < n) p[i] = 0.0f;
}

// ---------------------------------------------------------------------------
// K1: segment sum + counts.  batch is sorted, so run-length compress the
// atomics: each block handles 256 consecutive rows, threads own one column.
// x is streamed with non-temporal loads (read once, keep L2 for accumulators).
// ---------------------------------------------------------------------------
#define ROWS_PER_BLOCK 256
__global__ __launch_bounds__(128)
void seg_reduce_kernel(const float* __restrict__ x, const int* __restrict__ batch,
                       float* __restrict__ seg_sum, float* __restrict__ counts, int n) {
    __shared__ int sb[ROWS_PER_BLOCK];
    const int r0  = blockIdx.x * ROWS_PER_BLOCK;
    const int tid = threadIdx.x;                 // column 0..127
    const int nrows = min(ROWS_PER_BLOCK, n - r0);
    for (int i = tid; i < nrows; i += 128) sb[i] = batch[r0 + i];
    __syncthreads();

    float acc = 0.0f;
    int   cur = sb[0];
    for (int r = 0; r < nrows; ++r) {
        int s = sb[r];
        if (s != cur) {
            atomicAdd(&seg_sum[cur * H + tid], acc);
            acc = 0.0f;
            cur = s;
        }
        acc += __builtin_nontemporal_load(&x[(long)(r0 + r) * H + tid]);
    }
    atomicAdd(&seg_sum[cur * H + tid], acc);

    if (tid == 0) {  // run-length counts
        int c = sb[0], run = 1;
        for (int r = 1; r < nrows; ++r) {
            if (sb[r] == c) { run++; }
            else { atomicAdd(&counts[c], (float)run); c = sb[r]; run = 1; }
        }
        atomicAdd(&counts[c], (float)run);
    }
}

// ---------------------------------------------------------------------------
// WMMA f32 16x16x4 fragment conventions (ISA 7.12.2, 32-bit layouts):
//   lane = half*16 + idx  (half = lane>>4, idx = lane&15)
//   A(16x4):  a[j] = A[row=idx][k = 2*half + j]
//   B(4x16):  b[j] = B[k = 2*half + j][col=idx]
//   C/D:      acc[v] = C[row = v + 8*half][col=idx]
// ---------------------------------------------------------------------------

// K2/K4: Out[B,256] = relu( rowscale(A[B,128]) @ W[128,256] + bias[256] )
// rowscale = 1/max(counts,1) when counts != nullptr (encoder path), else 1.
__global__ __launch_bounds__(32)
void mlp_layer1_kernel(const float* __restrict__ A, const float* __restrict__ counts,
                       const float* __restrict__ W, const float* __restrict__ bias,
                       float* __restrict__ Out, int Bdim) {
    const int mt   = blockIdx.y;
    const int nt   = blockIdx.x;
    const int lane = threadIdx.x;
    const int idx  = lane & 15;
    const int half = lane >> 4;
    const int row  = min(mt * 16 + idx, Bdim - 1);   // clamp loads; EXEC stays full
    const int col  = nt * 16 + idx;

    float scale = 1.0f;
    if (counts) scale = 1.0f / fmaxf(counts[row], 1.0f);

    v8f acc = {};
    #pragma unroll 4
    for (int k = 0; k < H; k += 4) {
        const int ka = k + 2 * half;                   // even -> 8B aligned
        v2f a = *(const v2f*)(A + row * H + ka);
        a.x *= scale; a.y *= scale;
        v2f b;
        b.x = W[(ka + 0) * H2 + col];
        b.y = W[(ka + 1) * H2 + col];
        acc = __builtin_amdgcn_wmma_f32_16x16x4_f32(false, a, false, b,
                                                    (short)0, acc, false, false);
    }

    const float bv = bias[col];
    if (mt * 16 + 16 <= Bdim) {                        // uniform full-tile fast path
        float* outp = Out + (mt * 16 + 8 * half) * H2 + col;
        #pragma unroll
        for (int v = 0; v < 8; ++v)
            outp[v * H2] = fmaxf(acc[v] + bv, 0.0f);
    } else {
        #pragma unroll
        for (int v = 0; v < 8; ++v) {
            int m = mt * 16 + v + 8 * half;
            if (m < Bdim) Out[m * H2 + col] = fmaxf(acc[v] + bv, 0.0f);
        }
    }
}

// K3/K5: Out[B,128] = Hin[B,256] @ W[256,128] + bias[128] (+ addend[B,128])
// Optionally mirrors the result into Out2 (d_out tail for vn).
__global__ __launch_bounds__(32)
void mlp_layer2_kernel(const float* __restrict__ Hin, const float* __restrict__ W,
                       const float* __restrict__ bias, const float* __restrict__ addend,
                       float* __restrict__ Out, float* __restrict__ Out2, int Bdim) {
    const int mt   = blockIdx.y;
    const int nt   = blockIdx.x;
    const int lane = threadIdx.x;
    const int idx  = lane & 15;
    const int half = lane >> 4;
    const int row  = min(mt * 16 + idx, Bdim - 1);
    const int col  = nt * 16 + idx;

    v8f acc = {};
    #pragma unroll 4
    for (int k = 0; k < H2; k += 4) {
        const int ka = k + 2 * half;
        v2f a = *(const v2f*)(Hin + row * H2 + ka);
        v2f b;
        b.x = W[(ka + 0) * H + col];
        b.y = W[(ka + 1) * H + col];
        acc = __builtin_amdgcn_wmma_f32_16x16x4_f32(false, a, false, b,
                                                    (short)0, acc, false, false);
    }

    const float bv = bias[col];
    if (mt * 16 + 16 <= Bdim) {                        // uniform full-tile fast path
        const int m0 = mt * 16 + 8 * half;
        #pragma unroll
        for (int v = 0; v < 8; ++v) {
            int m = m0 + v;
            float r = acc[v] + bv;
            if (addend) r += addend[m * H + col];
            Out[m * H + col] = r;
            if (Out2) Out2[m * H + col] = r;
        }
    } else {
        #pragma unroll
        for (int v = 0; v < 8; ++v) {
            int m = mt * 16 + v + 8 * half;
            if (m < Bdim) {
                float r = acc[v] + bv;
                if (addend) r += addend[m * H + col];
                Out[m * H + col] = r;
                if (Out2) Out2[m * H + col] = r;
            }
        }
    }
}

// ---------------------------------------------------------------------------
// K6: x_out[i,:] = x[i,:] + dec_out[batch[i],:]   (pure streaming, 128-bit)
// x / x_out streamed non-temporally; dec_out (256KB) + batch stay L2-resident.
// ---------------------------------------------------------------------------
__global__ __launch_bounds__(256)
void broadcast_add_kernel(const v4f* __restrict__ x4, const int* __restrict__ batch,
                          const v4f* __restrict__ dec4, v4f* __restrict__ out4,
                          int total4) {
    int i = blockIdx.x * 256 + threadIdx.x;
    if (i >= total4) return;
    int row = i >> 5;          // H/4 = 32 vec4 per row
    int v   = i & 31;
    int b   = batch[row];
    v4f xv = __builtin_nontemporal_load(&x4[i]);
    v4f dv = dec4[b * 32 + v];
    v4f o  = xv + dv;
    __builtin_nontemporal_store(o, &out4[i]);
}

// ---------------------------------------------------------------------------
extern "C" void kernel_launch(void* const* d_in, const int* in_sizes, int n_in,
                              void* d_out, int out_size, void* d_ws, size_t ws_size,
                              hipStream_t stream) {
    const float* x      = (const float*)d_in[0];
    const float* vn_emb = (const float*)d_in[1];
    const int*   batch  = (const int*)d_in[2];
    const float* enc_w1 = (const float*)d_in[3];
    const float* enc_b1 = (const float*)d_in[4];
    const float* enc_w2 = (const float*)d_in[5];
    const float* enc_b2 = (const float*)d_in[6];
    const float* dec_w1 = (const float*)d_in[7];
    const float* dec_b1 = (const float*)d_in[8];
    const float* dec_w2 = (const float*)d_in[9];
    const float* dec_b2 = (const float*)d_in[10];

    const int N = in_sizes[0] / H;   // 500000
    const int B = in_sizes[1] / H;   // 512

    // workspace layout (floats)
    float* ws       = (float*)d_ws;
    float* seg_sum  = ws;                       // [B, H]
    float* counts   = seg_sum + (size_t)B * H;  // [B]
    float* h1       = counts + B;               // [B, 2H]
    float* vn       = h1 + (size_t)B * H2;      // [B, H]
    float* dec_out  = vn + (size_t)B * H;       // [B, H]

    float* x_out  = (float*)d_out;              // [N, H]
    float* vn_out = x_out + (size_t)N * H;      // [B, H]

    const int mtiles = (B + 15) / 16;

    // K0: zero accumulators
    int nzero = B * H + B;
    zero_kernel<<<(nzero + 255) / 256, 256, 0, stream>>>(seg_sum, nzero);

    // K1: segment reduce
    seg_reduce_kernel<<<(N + ROWS_PER_BLOCK - 1) / ROWS_PER_BLOCK, 128, 0, stream>>>(
        x, batch, seg_sum, counts, N);

    // K2: enc layer1 (per-row 1/count scaling folded into A loads)
    mlp_layer1_kernel<<<dim3(H2 / 16, mtiles), 32, 0, stream>>>(
        seg_sum, counts, enc_w1, enc_b1, h1, B);

    // K3: enc layer2, add vn_emb; write vn to ws and to d_out tail
    mlp_layer2_kernel<<<dim3(H / 16, mtiles), 32, 0, stream>>>(
        h1, enc_w2, enc_b2, vn_emb, vn, vn_out, B);

    // K4: dec layer1 (no row scaling)
    mlp_layer1_kernel<<<dim3(H2 / 16, mtiles), 32, 0, stream>>>(
        vn, (const float*)nullptr, dec_w1, dec_b1, h1, B);

    // K5: dec layer2 (no addend; final x add happens in streaming kernel)
    mlp_layer2_kernel<<<dim3(H / 16, mtiles), 32, 0, stream>>>(
        h1, dec_w2, dec_b2, (const float*)nullptr, dec_out, (float*)nullptr, B);

    // K6: x_out = x + dec_out[batch]
    int total4 = N * (H / 4);
    broadcast_add_kernel<<<(total4 + 255) / 256, 256, 0, stream>>>(
        (const v4f*)x, batch, (const v4f*)dec_out, (v4f*)x_out, total4);
}